// GATLayer_2662879724269
// MI455X (gfx1250) — compile-verified
//
#include <hip/hip_runtime.h>
#include <cmath>

typedef __attribute__((ext_vector_type(2))) float v2f;
typedef __attribute__((ext_vector_type(8))) float v8f;

#define HC    128      // HEADS*OUT_CH == IN_CH
#define HEADS 4
#define LDSW  132      // padded LDS stride (dwords) to kill bank conflicts

__device__ __forceinline__ unsigned f2mono(float f) {
  unsigned u = __float_as_uint(f);
  return (u & 0x80000000u) ? ~u : (u | 0x80000000u);
}
__device__ __forceinline__ float mono2f(unsigned u) {
  return (u & 0x80000000u) ? __uint_as_float(u ^ 0x80000000u)
                           : __uint_as_float(~u);
}
__device__ __forceinline__ float lrelu(float v) { return v > 0.f ? v : 0.2f * v; }

// ---------------------------------------------------------------------------
// 0) init: zero accumulator (d_out), denom; m = -inf (monotone-mapped)
// ---------------------------------------------------------------------------
__global__ void init_kernel(float* __restrict__ out, unsigned* __restrict__ m_bits,
                            float* __restrict__ denom, int N) {
  int i = blockIdx.x * blockDim.x + threadIdx.x;
  if (i < N * HC) out[i] = 0.f;
  if (i < N * HEADS) { m_bits[i] = f2mono(-__builtin_inff()); denom[i] = 0.f; }
}

// ---------------------------------------------------------------------------
// 1) h = x @ W  via V_WMMA_F32_16X16X4_F32.
//    Block = 256 threads (8 waves). Block computes 16 rows x 128 cols;
//    each wave owns one 16x16 tile. W + x-strip staged in LDS.
// ---------------------------------------------------------------------------
__global__ __launch_bounds__(256) void gemm_wmma(const float* __restrict__ x,
                                                 const float* __restrict__ W,
                                                 float* __restrict__ h) {
  __shared__ float Ws[128 * LDSW];
  __shared__ float xs[16 * LDSW];
  const int tid = threadIdx.x;
  const int m0  = blockIdx.x * 16;

  for (int i = tid; i < 128 * 128; i += 256)
    Ws[(i >> 7) * LDSW + (i & 127)] = W[i];
  for (int i = tid; i < 16 * 128; i += 256) {
    int r = i >> 7, c = i & 127;
    xs[r * LDSW + c] = x[(m0 + r) * HC + c];
  }
  __syncthreads();

  const int wave = tid >> 5, lane = tid & 31;
  const int n0 = wave * 16;
  const int m  = lane & 15;       // row (A) / col (B) within tile
  const int kg = lane >> 4;       // K half-select

  v8f c = {};
#pragma unroll
  for (int k0 = 0; k0 < 128; k0 += 4) {
    const int ka = k0 + 2 * kg;
    v2f a, b;
    a.x = xs[m * LDSW + ka];
    a.y = xs[m * LDSW + ka + 1];
    b.x = Ws[ka * LDSW + n0 + m];
    b.y = Ws[(ka + 1) * LDSW + n0 + m];
    c = __builtin_amdgcn_wmma_f32_16x16x4_f32(false, a, false, b, (short)0, c,
                                              false, false);
  }
#pragma unroll
  for (int r = 0; r < 8; ++r)
    h[(m0 + r + 8 * kg) * HC + n0 + m] = c[r];
}

// ---------------------------------------------------------------------------
// 2) per-node attention logits: a_src[n,h] = sum_c h[n,h,c]*att_src[h,c]
//    128 threads/node; each head == one wave32 -> shfl_xor reduction.
// ---------------------------------------------------------------------------
__global__ __launch_bounds__(128) void att_kernel(const float* __restrict__ h,
                                                  const float* __restrict__ att_src,
                                                  const float* __restrict__ att_dst,
                                                  float* __restrict__ a_src,
                                                  float* __restrict__ a_dst) {
  const int n = blockIdx.x;
  const int t = threadIdx.x;
  const int head = t >> 5, lane = t & 31;
  const float hv = h[n * HC + t];
  float s = hv * att_src[t];
  float d = hv * att_dst[t];
#pragma unroll
  for (int off = 16; off; off >>= 1) {
    s += __shfl_xor(s, off);
    d += __shfl_xor(d, off);
  }
  if (lane == 0) { a_src[n * HEADS + head] = s; a_dst[n * HEADS + head] = d; }
}

// ---------------------------------------------------------------------------
// 3) segment max of leaky_relu(a_src[src]+a_dst[dst]) into m_bits (uint max)
// ---------------------------------------------------------------------------
__global__ void edge_max(const int* __restrict__ ei, const float* __restrict__ a_src,
                         const float* __restrict__ a_dst, unsigned* __restrict__ m_bits,
                         int E, int N) {
  const int idx = blockIdx.x * blockDim.x + threadIdx.x;
  if (idx >= E + N) return;
  int s, d;
  if (idx < E) { s = ei[idx]; d = ei[E + idx]; } else { s = d = idx - E; }
  const float4 as = *(const float4*)(a_src + s * HEADS);
  const float4 ad = *(const float4*)(a_dst + d * HEADS);
  const float e[4] = {as.x + ad.x, as.y + ad.y, as.z + ad.z, as.w + ad.w};
#pragma unroll
  for (int hh = 0; hh < HEADS; ++hh)
    atomicMax(&m_bits[d * HEADS + hh], f2mono(lrelu(e[hh])));
}

// ---------------------------------------------------------------------------
// 4) denom[dst,h] += exp(e - m[dst,h])   (native f32 atomic add)
// ---------------------------------------------------------------------------
__global__ void edge_sum(const int* __restrict__ ei, const float* __restrict__ a_src,
                         const float* __restrict__ a_dst,
                         const unsigned* __restrict__ m_bits,
                         float* __restrict__ denom, int E, int N) {
  const int idx = blockIdx.x * blockDim.x + threadIdx.x;
  if (idx >= E + N) return;
  int s, d;
  if (idx < E) { s = ei[idx]; d = ei[E + idx]; } else { s = d = idx - E; }
  const float4 as = *(const float4*)(a_src + s * HEADS);
  const float4 ad = *(const float4*)(a_dst + d * HEADS);
  const float e[4] = {as.x + ad.x, as.y + ad.y, as.z + ad.z, as.w + ad.w};
#pragma unroll
  for (int hh = 0; hh < HEADS; ++hh) {
    const float p = __expf(lrelu(e[hh]) - mono2f(m_bits[d * HEADS + hh]));
    unsafeAtomicAdd(&denom[d * HEADS + hh], p);
  }
}

// ---------------------------------------------------------------------------
// 5) scatter: out[dst] += alpha * h[src]. One wave32 per edge; each lane
//    moves 4 channels (float4 gather + 4x global_atomic_add_f32).
// ---------------------------------------------------------------------------
__global__ __launch_bounds__(256) void edge_scatter(
    const int* __restrict__ ei, const float* __restrict__ h,
    const float* __restrict__ a_src, const float* __restrict__ a_dst,
    const unsigned* __restrict__ m_bits, const float* __restrict__ denom,
    float* __restrict__ out, int E, int N) {
  const int edge = blockIdx.x * 8 + (threadIdx.x >> 5);
  if (edge >= E + N) return;
  const int lane = threadIdx.x & 31;
  int s, d;
  if (edge < E) { s = ei[edge]; d = ei[E + edge]; } else { s = d = edge - E; }

  float al = 0.f;
  if (lane < HEADS) {
    const float e = lrelu(a_src[s * HEADS + lane] + a_dst[d * HEADS + lane]);
    const float p = __expf(e - mono2f(m_bits[d * HEADS + lane]));
    al = p / (denom[d * HEADS + lane] + 1e-16f);
  }
  const float a = __shfl(al, lane >> 3);   // head = lane/8 (4 ch per lane)

  const float4 hv = *(const float4*)(h + s * HC + lane * 4);
  float* o = out + d * HC + lane * 4;
  unsafeAtomicAdd(o + 0, a * hv.x);
  unsafeAtomicAdd(o + 1, a * hv.y);
  unsafeAtomicAdd(o + 2, a * hv.z);
  unsafeAtomicAdd(o + 3, a * hv.w);
}

// ---------------------------------------------------------------------------
// 6) bias + ELU + residual + LayerNorm. One wave32 per node, 4 ch/lane.
// ---------------------------------------------------------------------------
__global__ __launch_bounds__(256) void finalize(float* __restrict__ out,
                                                const float* __restrict__ x,
                                                const float* __restrict__ bias,
                                                const float* __restrict__ gamma,
                                                const float* __restrict__ beta,
                                                int N) {
  const int node = blockIdx.x * 8 + (threadIdx.x >> 5);
  if (node >= N) return;
  const int lane = threadIdx.x & 31;
  const int c0 = lane * 4;

  const float4 v  = *(const float4*)(out + node * HC + c0);
  const float4 xr = *(const float4*)(x + node * HC + c0);
  const float vi[4] = {v.x, v.y, v.z, v.w};
  const float xi[4] = {xr.x, xr.y, xr.z, xr.w};
  float r[4], s = 0.f, s2 = 0.f;
#pragma unroll
  for (int i = 0; i < 4; ++i) {
    const float val = vi[i] + bias[c0 + i];
    const float e = val > 0.f ? val : expm1f(val);
    r[i] = e + xi[i];
    s += r[i];
    s2 += r[i] * r[i];
  }
#pragma unroll
  for (int off = 16; off; off >>= 1) {
    s  += __shfl_xor(s, off);
    s2 += __shfl_xor(s2, off);
  }
  const float mean = s * (1.f / HC);
  const float var  = s2 * (1.f / HC) - mean * mean;
  const float inv  = rsqrtf(var + 1e-5f);
  float4 o;
  o.x = (r[0] - mean) * inv * gamma[c0 + 0] + beta[c0 + 0];
  o.y = (r[1] - mean) * inv * gamma[c0 + 1] + beta[c0 + 1];
  o.z = (r[2] - mean) * inv * gamma[c0 + 2] + beta[c0 + 2];
  o.w = (r[3] - mean) * inv * gamma[c0 + 3] + beta[c0 + 3];
  *(float4*)(out + node * HC + c0) = o;
}

// ---------------------------------------------------------------------------
extern "C" void kernel_launch(void* const* d_in, const int* in_sizes, int n_in,
                              void* d_out, int out_size, void* d_ws, size_t ws_size,
                              hipStream_t stream) {
  const float* x        = (const float*)d_in[0];
  const int*   ei       = (const int*)d_in[1];
  const float* W        = (const float*)d_in[2];
  const float* att_src  = (const float*)d_in[3];
  const float* att_dst  = (const float*)d_in[4];
  const float* bias     = (const float*)d_in[5];
  const float* gamma    = (const float*)d_in[6];
  const float* beta     = (const float*)d_in[7];
  float* out = (float*)d_out;

  const int N = in_sizes[0] / HC;      // 50000
  const int E = in_sizes[1] / 2;       // 800000
  const int EL = E + N;                // edges incl. self loops

  char* ws = (char*)d_ws;
  float*    h      = (float*)ws;                       ws += (size_t)N * HC * 4;
  float*    a_src  = (float*)ws;                       ws += (size_t)N * HEADS * 4;
  float*    a_dst  = (float*)ws;                       ws += (size_t)N * HEADS * 4;
  unsigned* m_bits = (unsigned*)ws;                    ws += (size_t)N * HEADS * 4;
  float*    denom  = (float*)ws;

  init_kernel<<<(N * HC + 255) / 256, 256, 0, stream>>>(out, m_bits, denom, N);
  gemm_wmma<<<N / 16, 256, 0, stream>>>(x, W, h);
  att_kernel<<<N, 128, 0, stream>>>(h, att_src, att_dst, a_src, a_dst);
  edge_max<<<(EL + 255) / 256, 256, 0, stream>>>(ei, a_src, a_dst, m_bits, E, N);
  edge_sum<<<(EL + 255) / 256, 256, 0, stream>>>(ei, a_src, a_dst, m_bits, denom, E, N);
  edge_scatter<<<(EL + 7) / 8, 256, 0, stream>>>(ei, h, a_src, a_dst, m_bits, denom,
                                                 out, E, N);
  finalize<<<(N + 7) / 8, 256, 0, stream>>>(out, x, bias, gamma, beta, N);
}